// ROILoss_23450521436767
// MI455X (gfx1250) — compile-verified
//
#include <hip/hip_runtime.h>
#include <hip/hip_bf16.h>
#include <math.h>

// Problem constants (match reference)
#define BATCH 32
#define RROWS 4096
#define NCLS  81
#define C4    324          // 4 * NCLS
#define NQUAD 81           // C4 / 4 float4 quads per bbox row
#define CHUNK 64           // rows per block in the streaming kernel
#define NCHUNK (RROWS / CHUNK)   // 64

typedef __attribute__((ext_vector_type(2))) float v2f;
typedef __attribute__((ext_vector_type(8))) float v8f;

// ---------------------------------------------------------------------------
// Wave32 all-reduce (sum) on the matrix pipe.
// Stage 1: A = [v | 0] (16x4), B = ones(4x16)  ->  D[m][n] = v[m] + v[m+16]
// Stages 2-5: feed D's 8 VGPRs back as four 16x4 A-matrices, accumulate into
// one C. Every element of the final accumulator holds sum(v[0..31]).
// EXEC must be all ones at the call site (called from reconverged code).
// ---------------------------------------------------------------------------
__device__ __forceinline__ float wave_allreduce_sum_wmma(float v) {
  v2f a; a[0] = v; a[1] = 0.0f;
  v2f ones; ones[0] = 1.0f; ones[1] = 1.0f;
  v8f zero = {};
  v8f d1 = __builtin_amdgcn_wmma_f32_16x16x4_f32(false, a, false, ones,
                                                 (short)0, zero, false, false);
  v8f acc = {};
  v2f p;
  p[0] = d1[0]; p[1] = d1[1];
  acc = __builtin_amdgcn_wmma_f32_16x16x4_f32(false, p, false, ones,
                                              (short)0, acc, false, false);
  p[0] = d1[2]; p[1] = d1[3];
  acc = __builtin_amdgcn_wmma_f32_16x16x4_f32(false, p, false, ones,
                                              (short)0, acc, false, false);
  p[0] = d1[4]; p[1] = d1[5];
  acc = __builtin_amdgcn_wmma_f32_16x16x4_f32(false, p, false, ones,
                                              (short)0, acc, false, false);
  p[0] = d1[6]; p[1] = d1[7];
  acc = __builtin_amdgcn_wmma_f32_16x16x4_f32(false, p, false, ones,
                                              (short)0, acc, false, false);
  return acc[0];
}

__device__ __forceinline__ float smooth_l1(float d) {
  const float ad = fabsf(d);
  return (ad < 1.0f) ? 0.5f * d * d : (ad - 0.5f);
}

// ---------------------------------------------------------------------------
// Kernel 1: per-image scan of fg flags.
//   chunk_base[b][c] = #foreground rows before row c*CHUNK of image b
//   nfg[b]           = total foreground rows in image b
// Also zero-initializes the per-image accumulators (deterministic per call).
// Grid: BATCH blocks x 256 threads.
// ---------------------------------------------------------------------------
__global__ void roiloss_scan_kernel(const int* __restrict__ labels,
                                    int* __restrict__ chunk_base,
                                    int* __restrict__ nfg,
                                    float* __restrict__ cls_sum,
                                    float* __restrict__ bbox_sum) {
  const int b = blockIdx.x;
  const int t = threadIdx.x;           // 0..255, each owns 16 rows
  const int* lb = labels + (size_t)b * RROWS + t * 16;
  int cnt = 0;
#pragma unroll
  for (int k = 0; k < 16; ++k) cnt += (lb[k] > 0) ? 1 : 0;

  __shared__ int s[256];
  s[t] = cnt;
  __syncthreads();
  // Hillis-Steele inclusive scan over 256 counts
  for (int off = 1; off < 256; off <<= 1) {
    int add = (t >= off) ? s[t - off] : 0;
    __syncthreads();
    s[t] += add;
    __syncthreads();
  }
  // Chunk c covers rows [c*64, c*64+64) == threads [4c, 4c+4)
  if ((t & 3) == 0) {
    int c = t >> 2;
    chunk_base[b * NCHUNK + c] = (t == 0) ? 0 : s[t - 1];
  }
  if (t == 0) {
    nfg[b] = s[255];
    cls_sum[b] = 0.0f;
    bbox_sum[b] = 0.0f;
  }
}

// ---------------------------------------------------------------------------
// Kernel 2: streaming loss kernel. Wave-per-row, 8 waves per block,
// 64 rows per block. Grid: (NCHUNK, BATCH) x 256 threads.
// ---------------------------------------------------------------------------
__global__ void __launch_bounds__(256)
roiloss_main_kernel(const float* __restrict__ cls_logits,
                    const float* __restrict__ bbox_preds,
                    const int* __restrict__ labels,
                    const float* __restrict__ bbox_deltas,
                    const int* __restrict__ chunk_base,
                    float* __restrict__ cls_sum,
                    float* __restrict__ bbox_sum) {
  const int chunk = blockIdx.x;
  const int b = blockIdx.y;
  const int row0 = chunk * CHUNK;
  const int tid = threadIdx.x;
  const int wave = tid >> 5;
  const int lane = tid & 31;

  __shared__ int s_lab[CHUNK];
  __shared__ int s_rank[CHUNK];
  __shared__ float s_sl1[8];
  __shared__ float s_nll[8];

  if (tid < CHUNK)
    s_lab[tid] = labels[(size_t)b * RROWS + row0 + tid];
  __syncthreads();
  if (tid == 0) {
    // exclusive prefix of fg within the chunk, offset by the image-wide base
    int run = chunk_base[b * NCHUNK + chunk];
    for (int j = 0; j < CHUNK; ++j) {
      s_rank[j] = run;
      run += (s_lab[j] > 0) ? 1 : 0;
    }
  }
  __syncthreads();

  float nll_acc = 0.0f;   // only lane 0 accumulates (value is lane-uniform)
  float sl1_acc = 0.0f;   // per-lane partial

  for (int j = wave; j < CHUNK; j += 8) {
    const int r = row0 + j;
    const int label = s_lab[j];
    const size_t row_id = (size_t)b * RROWS + r;

    // ---- cross entropy: log-sum-exp over 81 logits, wave-cooperative ----
    const float* xr = cls_logits + row_id * NCLS;
    const bool v2ok = (lane < (NCLS - 64));      // lane < 17
    float v0 = xr[lane];
    float v1 = xr[lane + 32];
    float v2 = v2ok ? xr[lane + 64] : -INFINITY;

    float mx = fmaxf(fmaxf(v0, v1), v2);
#pragma unroll
    for (int m = 16; m >= 1; m >>= 1) mx = fmaxf(mx, __shfl_xor(mx, m, 32));

    float se = expf(v0 - mx) + expf(v1 - mx) + (v2ok ? expf(v2 - mx) : 0.0f);
#pragma unroll
    for (int m = 16; m >= 1; m >>= 1) se += __shfl_xor(se, m, 32);

    float lse = mx + logf(se);
    float xl = xr[label];                        // uniform address: broadcast
    float nll = lse - xl;
    nll_acc += (lane == 0) ? nll : 0.0f;

    // ---- smooth L1 over 324 columns (foreground rows only; wave-uniform) --
    // Row = 81 aligned float4 quads; the target slot label*4..label*4+3 is
    // exactly quad q == label, so each quad needs one compare, not four.
    if (label > 0) {
      const float4* pr4 = (const float4*)(bbox_preds + row_id * C4);
      // prefetch the next row this wave will stream (8 rows ahead)
      if (j + 8 < CHUNK) {
        const float* nxt = bbox_preds + (row_id + 8) * C4;
        __builtin_prefetch(nxt + lane * 16, 0, 0);
      }
      const int rk = s_rank[j];                  // valid in [0, nfg) for fg
      const float4 dv =
          *(const float4*)(bbox_deltas + ((size_t)b * RROWS + rk) * 4);
#pragma unroll
      for (int s = 0; s < 3; ++s) {
        const int q = lane + s * 32;             // quad index in [0, 81)
        if (q < NQUAD) {
          const float4 pv = pr4[q];              // global_load_b128
          float tx = 0.0f, ty = 0.0f, tz = 0.0f, tw = 0.0f;
          if (q == label) { tx = dv.x; ty = dv.y; tz = dv.z; tw = dv.w; }
          sl1_acc += smooth_l1(pv.x - tx) + smooth_l1(pv.y - ty) +
                     smooth_l1(pv.z - tz) + smooth_l1(pv.w - tw);
        }
      }
    }
  }

  // ---- per-wave all-reduce on the matrix pipe (EXEC reconverged here) ----
  const float tot_sl1 = wave_allreduce_sum_wmma(sl1_acc);
  const float tot_nll = wave_allreduce_sum_wmma(nll_acc);
  if (lane == 0) {
    s_sl1[wave] = tot_sl1;
    s_nll[wave] = tot_nll;
  }
  __syncthreads();
  if (tid == 0) {
    float a_sl1 = 0.0f, a_nll = 0.0f;
#pragma unroll
    for (int w = 0; w < 8; ++w) { a_sl1 += s_sl1[w]; a_nll += s_nll[w]; }
    atomicAdd(&bbox_sum[b], a_sl1);
    atomicAdd(&cls_sum[b], a_nll);
  }
}

// ---------------------------------------------------------------------------
// Kernel 3: finalize. One wave; per-image roi, masked by nfg>0, mean over B.
// ---------------------------------------------------------------------------
__global__ void roiloss_finalize_kernel(const float* __restrict__ cls_sum,
                                        const float* __restrict__ bbox_sum,
                                        const int* __restrict__ nfg,
                                        float* __restrict__ out) {
  const int t = threadIdx.x;  // 32 threads
  float roi = 0.0f;
  if (t < BATCH) {
    const int nf = nfg[t];
    const float cls_loss = cls_sum[t] / (float)RROWS;
    const float denom = fmaxf((float)nf * (float)C4, 1.0f);
    const float bbox_loss = bbox_sum[t] / denom;
    roi = (nf > 0) ? (cls_loss + bbox_loss) : 0.0f;
  }
#pragma unroll
  for (int m = 16; m >= 1; m >>= 1) roi += __shfl_xor(roi, m, 32);
  if (t == 0) out[0] = roi / (float)BATCH;
}

// ---------------------------------------------------------------------------
extern "C" void kernel_launch(void* const* d_in, const int* in_sizes, int n_in,
                              void* d_out, int out_size, void* d_ws, size_t ws_size,
                              hipStream_t stream) {
  const float* cls_logits  = (const float*)d_in[0];  // [B,R,81]
  const float* bbox_preds  = (const float*)d_in[1];  // [B,R,324]
  const int*   labels      = (const int*)d_in[2];    // [B,R]
  const float* bbox_deltas = (const float*)d_in[3];  // [B,R,4]
  float* out = (float*)d_out;

  // Workspace layout (all within ~8.5 KB)
  float* cls_sum   = (float*)d_ws;                   // [B]
  float* bbox_sum  = cls_sum + BATCH;                // [B]
  int*   nfg       = (int*)(bbox_sum + BATCH);       // [B]
  int*   chunk_base = nfg + BATCH;                   // [B*NCHUNK]

  roiloss_scan_kernel<<<BATCH, 256, 0, stream>>>(labels, chunk_base, nfg,
                                                 cls_sum, bbox_sum);

  dim3 grid(NCHUNK, BATCH);
  roiloss_main_kernel<<<grid, 256, 0, stream>>>(cls_logits, bbox_preds, labels,
                                                bbox_deltas, chunk_base,
                                                cls_sum, bbox_sum);

  roiloss_finalize_kernel<<<1, 32, 0, stream>>>(cls_sum, bbox_sum, nfg, out);
}